// GeodesicConv_70497593196912
// MI455X (gfx1250) — compile-verified
//
#include <hip/hip_runtime.h>
#include <hip/hip_bf16.h>

#define N_RINGS   5
#define N_ORIENT  8
#define NBINS     (N_RINGS * N_ORIENT)
#define CIN       32
#define COUT      32
#define TWO_PI_F  6.28318f

typedef __attribute__((ext_vector_type(2))) float v2f;
typedef __attribute__((ext_vector_type(8))) float v8f;

__device__ __forceinline__ int point_bin(const float* __restrict__ coords, int i) {
    float radial  = coords[2 * i + 0];
    float angular = coords[2 * i + 1];
    int ring = (int)(radial * (float)N_RINGS);
    ring = ring < 0 ? 0 : (ring > N_RINGS - 1 ? N_RINGS - 1 : ring);
    int orient = (int)(angular * ((float)N_ORIENT / TWO_PI_F));
    orient = orient < 0 ? 0 : (orient > N_ORIENT - 1 ? N_ORIENT - 1 : orient);
    return ring * N_ORIENT + orient;
}

// ---------------- Pass 0: init counters + padded index list ----------------
__global__ void gc_init(int* __restrict__ binCount, int* __restrict__ fillPos,
                        int* __restrict__ pointIdx, int maxPadded) {
    int i = blockIdx.x * blockDim.x + threadIdx.x;
    if (i < NBINS) { binCount[i] = 0; fillPos[i] = 0; }
    for (int j = i; j < maxPadded; j += gridDim.x * blockDim.x)
        pointIdx[j] = -1;
}

// ---------------- Pass 1: histogram points into 40 bins --------------------
__global__ void gc_count(const float* __restrict__ coords, int* __restrict__ binCount,
                         int total) {
    int i = blockIdx.x * blockDim.x + threadIdx.x;
    if (i >= total) return;
    atomicAdd(&binCount[point_bin(coords, i)], 1);
}

// ---------------- Pass 2: exclusive scan, bins padded to 16 ----------------
__global__ void gc_scan(const int* __restrict__ binCount, int* __restrict__ binStart) {
    if (threadIdx.x == 0 && blockIdx.x == 0) {
        int acc = 0;
        for (int b = 0; b < NBINS; ++b) {
            binStart[b] = acc;
            acc += (binCount[b] + 15) & ~15;   // pad each bin to a multiple of 16
        }
        binStart[NBINS] = acc;
    }
}

// ---------------- Pass 3: scatter point ids into sorted list ---------------
__global__ void gc_scatter(const float* __restrict__ coords,
                           const int* __restrict__ binStart, int* __restrict__ fillPos,
                           int* __restrict__ pointIdx, int total) {
    int i = blockIdx.x * blockDim.x + threadIdx.x;
    if (i >= total) return;
    int bin = point_bin(coords, i);
    int pos = atomicAdd(&fillPos[bin], 1);
    pointIdx[binStart[bin] + pos] = i;
}

// ---------------- Pass 4: gathered WMMA GEMM -------------------------------
// One wave32 per 16-point tile. A = 16x32 gathered features, B = bin's 32x32 W,
// C/D = 16x32 output tile, computed as 8 K-chunks x 2 COUT halves of
// v_wmma_f32_16x16x4_f32 with bias preloaded into the accumulators.
__global__ void gc_gemm(const float* __restrict__ features,
                        const float* __restrict__ filters,
                        const float* __restrict__ bias,
                        const int* __restrict__ binStart,
                        const int* __restrict__ pointIdx,
                        float* __restrict__ out) {
    const int wave = threadIdx.x >> 5;                       // 8 waves / block
    const int tile = blockIdx.x * (blockDim.x >> 5) + wave;
    const int tileStart = tile * 16;
    const int totalPadded = binStart[NBINS];
    if (tileStart >= totalPadded) return;                    // wave-uniform exit

    // Which bin does this tile belong to? (tiles never straddle bins: padded)
    int bin = 0;
    #pragma unroll
    for (int b = 1; b < NBINS; ++b)
        if (tileStart >= binStart[b]) bin = b;

    const int lane = threadIdx.x & 31;
    const int col  = lane & 15;          // N for B/C/D; M row for A
    const int half = lane >> 4;          // 0: K={0,1}; 1: K={2,3} (A/B striping)
    const int kkBase = half * 2;

    // Gather row index (lanes 16-31 mirror rows 0-15 for the A fragment)
    const int p = pointIdx[tileStart + col];
    const float* __restrict__ frow = features + (long)(p < 0 ? 0 : p) * CIN;
    const float* __restrict__ W = filters + (long)bin * (CIN * COUT);

    // Accumulators start at bias (every element of c VGPR j has column `col`)
    v8f acc0, acc1;
    const float b0 = bias[col], b1 = bias[col + 16];
    #pragma unroll
    for (int j = 0; j < 8; ++j) { acc0[j] = b0; acc1[j] = b1; }

    #pragma unroll
    for (int k = 0; k < 8; ++k) {
        const int kb = 4 * k + kkBase;
        v2f a;                                   // A 16x4: v0=K kb, v1=K kb+1
        a.x = frow[kb];
        a.y = frow[kb + 1];

        v2f bm0;                                 // B 4x16, COUT half 0
        bm0.x = W[kb * COUT + col];
        bm0.y = W[(kb + 1) * COUT + col];
        acc0 = __builtin_amdgcn_wmma_f32_16x16x4_f32(
            false, a, false, bm0, (short)0, acc0, false, false);

        v2f bm1;                                 // B 4x16, COUT half 1
        bm1.x = W[kb * COUT + col + 16];
        bm1.y = W[(kb + 1) * COUT + col + 16];
        acc1 = __builtin_amdgcn_wmma_f32_16x16x4_f32(
            false, a, false, bm1, (short)0, acc1, false, false);
    }

    // C/D layout: VGPR j -> rows {j, j+8}, N = lane%16 (+16 for acc1)
    #pragma unroll
    for (int j = 0; j < 8; ++j) {
        const int m = j + half * 8;
        const int pm = __shfl(p, m, 32);
        if (pm >= 0) {
            out[(long)pm * COUT + col]      = acc0[j];
            out[(long)pm * COUT + col + 16] = acc1[j];
        }
    }
}

extern "C" void kernel_launch(void* const* d_in, const int* in_sizes, int n_in,
                              void* d_out, int out_size, void* d_ws, size_t ws_size,
                              hipStream_t stream) {
    const float* features = (const float*)d_in[0];   // (B,N,32) f32
    const float* coords   = (const float*)d_in[1];   // (B,N,2)  f32
    const float* filters  = (const float*)d_in[2];   // (5,8,32,32) f32
    const float* bias     = (const float*)d_in[3];   // (32,) f32
    float* out = (float*)d_out;

    const int totalPoints = in_sizes[1] / 2;         // B*N
    const int maxPadded   = totalPoints + NBINS * 16;
    const int maxTiles    = maxPadded / 16;

    // Workspace layout
    int* binCount = (int*)d_ws;                      // 40
    int* fillPos  = binCount + NBINS;                // 40
    int* binStart = fillPos + NBINS;                 // 41
    int* pointIdx = binStart + (NBINS + 1);          // maxPadded

    const int T = 256;
    gc_init   <<<(maxPadded + T - 1) / T, T, 0, stream>>>(binCount, fillPos, pointIdx, maxPadded);
    gc_count  <<<(totalPoints + T - 1) / T, T, 0, stream>>>(coords, binCount, totalPoints);
    gc_scan   <<<1, 1, 0, stream>>>(binCount, binStart);
    gc_scatter<<<(totalPoints + T - 1) / T, T, 0, stream>>>(coords, binStart, fillPos, pointIdx, totalPoints);

    const int wavesPerBlock = T / 32;                // 8 tiles per block
    gc_gemm   <<<(maxTiles + wavesPerBlock - 1) / wavesPerBlock, T, 0, stream>>>(
        features, filters, bias, binStart, pointIdx, out);
}